// EnhancedHGNNConv_35098472743557
// MI455X (gfx1250) — compile-verified
//
#include <hip/hip_runtime.h>
#include <hip/hip_bf16.h>

#define N_NODES 100000
#define N_EDGES 20000
#define NNZ     1600000
#define D_IN    256
#define D_OUT   128
#define HEADS   4
#define HEAD_DIM 32
#define LN_EPS  1e-5f

typedef __attribute__((ext_vector_type(2))) float v2f;
typedef __attribute__((ext_vector_type(8))) float v8f;

// ---------------------------------------------------------------------------
// GEMM: C[M, 128] = A[M, 256] @ W[256, 128] + bias, fp32 via V_WMMA_F32_16X16X4_F32
// Block = 256 threads = 8 waves. Each wave computes a 16-row x 128-col strip.
// W staged in LDS in two 64KB K-chunks, PAIR-INTERLEAVED so that each B
// fragment {W[k][n], W[k+1][n]} is one aligned ds_load_b64.
// A fragments are software-pipelined (next k-step prefetched before WMMA burst).
// ---------------------------------------------------------------------------
__global__ __launch_bounds__(256) void gemm_wmma_k(
    const float* __restrict__ A, const float* __restrict__ W,
    const float* __restrict__ bias, float* __restrict__ C, int M) {
  // sW[kpair][n][0..1] : element (2*kpair + j, n) of current K-chunk
  __shared__ float sW[64 * D_OUT * 2];  // 64 KB

  const int tid   = threadIdx.x;
  const int wave  = tid >> 5;                // 0..7
  const int lane  = tid & 31;
  const int laneN = lane & 15;
  const int khalf = (lane >> 4) << 1;        // 0 or 2 (K sub-pair per lane half)
  const int mhalf = (lane >> 4) << 3;        // 0 or 8 (C row half per lane half)

  const int rowBase = blockIdx.x * 128 + wave * 16;
  int rowA = rowBase + laneN;
  if (rowA > M - 1) rowA = M - 1;            // clamp (keep EXEC all-ones for WMMA)
  const float* __restrict__ Arow = A + (size_t)rowA * D_IN;

  v8f acc[8] = {};

  // initial A fragment (k = khalf)
  v2f a;
  a.x = Arow[khalf];
  a.y = Arow[khalf + 1];

  for (int kc = 0; kc < D_IN; kc += 128) {
    if (kc) __syncthreads();                 // protect LDS before overwrite
    // stage 128x128 K-chunk of W, pair-interleaved: 32 pairs per thread
    {
      const float* __restrict__ Wc = W + (size_t)kc * D_OUT;
      #pragma unroll
      for (int it = 0; it < 32; ++it) {
        const int p = tid + it * 256;        // pair index: 64 kpairs x 128 n
        const int kp = p >> 7;               // 0..63
        const int n  = p & 127;
        float2 pr;
        pr.x = Wc[(kp * 2 + 0) * D_OUT + n];
        pr.y = Wc[(kp * 2 + 1) * D_OUT + n];
        *(float2*)(sW + kp * 256 + n * 2) = pr;
      }
    }
    __syncthreads();

    for (int k0 = 0; k0 < 128; k0 += 4) {
      // prefetch next k-step's A fragment (wraps to k=0 at the end: valid addr)
      const int kg = (kc + k0 + 4) & (D_IN - 1);
      v2f an;
      an.x = Arow[kg + khalf];
      an.y = Arow[kg + khalf + 1];

      // B fragments: contiguous aligned pairs in interleaved LDS
      const float* __restrict__ bp = sW + ((k0 + khalf) >> 1) * 256 + laneN * 2;
      #pragma unroll
      for (int t = 0; t < 8; ++t) {
        const v2f b = *(const v2f*)(bp + t * 32);
        acc[t] = __builtin_amdgcn_wmma_f32_16x16x4_f32(
            false, a, false, b, (short)0, acc[t], false, false);
      }
      a = an;
    }
  }

  #pragma unroll
  for (int t = 0; t < 8; ++t) {
    const int col = t * 16 + laneN;
    const float bc = bias[col];
    #pragma unroll
    for (int r = 0; r < 8; ++r) {
      const int row = rowBase + mhalf + r;
      if (row < M) C[(size_t)row * D_OUT + col] = acc[t][r] + bc;
    }
  }
}

// ---------------------------------------------------------------------------
// Collapse proj+attn-vec:  V[k, o] so that score_o(row) = trans[row,:] @ V[:,o]
// node side (from Wnp):  o<4 -> qn (av_e2n[:, :32]),  o>=4 -> kn (av_n2e[:, 32:])
// edge side (from Whp):  o<4 -> ke (av_e2n[:, 32:]),  o>=4 -> qe (av_n2e[:, :32])
// ---------------------------------------------------------------------------
__global__ void build_attn_vecs_k(const float* __restrict__ Wnp,
                                  const float* __restrict__ Whp,
                                  const float* __restrict__ av_e2n,
                                  const float* __restrict__ av_n2e,
                                  float* __restrict__ Vn, float* __restrict__ Ve) {
  int t = blockIdx.x * blockDim.x + threadIdx.x;
  if (t >= 2 * 128 * 8) return;
  const int which = t / 1024;           // 0 = node side, 1 = edge side
  const int r = t % 1024;
  const int k = r / 8, o = r % 8, h = o & 3;
  const float* Wp = which ? Whp : Wnp;
  const float* av; int off;
  if (which == 0) { if (o < 4) { av = av_e2n; off = 0; } else { av = av_n2e; off = HEAD_DIM; } }
  else            { if (o < 4) { av = av_e2n; off = HEAD_DIM; } else { av = av_n2e; off = 0; } }
  float s = 0.f;
  #pragma unroll
  for (int d = 0; d < HEAD_DIM; ++d)
    s += Wp[k * 128 + h * HEAD_DIM + d] * av[h * 2 * HEAD_DIM + off + d];
  (which ? Ve : Vn)[k * 8 + o] = s;
}

// scores[row*8 + o] = trans[row,:] @ V[:,o]
__global__ __launch_bounds__(256) void scores_k(const float* __restrict__ trans,
                                                const float* __restrict__ V,
                                                float* __restrict__ out, int M) {
  __shared__ float sV[128 * 8];
  for (int i = threadIdx.x; i < 1024; i += 256) sV[i] = V[i];
  __syncthreads();
  const int t = blockIdx.x * 256 + threadIdx.x;
  if (t >= M * 8) return;
  const int row = t >> 3, o = t & 7;
  const float* __restrict__ r = trans + (size_t)row * D_OUT;
  float s = 0.f;
  #pragma unroll 8
  for (int k = 0; k < D_OUT; ++k) s += r[k] * sV[k * 8 + o];
  out[t] = s;
}

// ---------------------------------------------------------------------------
// Segment softmax + aggregation buffers init
// ---------------------------------------------------------------------------
__global__ void init_k(float* m_n, float* m_e, float* sum_n, float* sum_e,
                       float* x_agg, float* y_agg) {
  const size_t i = (size_t)blockIdx.x * blockDim.x + threadIdx.x;
  const size_t stride = (size_t)gridDim.x * blockDim.x;
  const float ninf = __int_as_float(0xff800000);
  for (size_t j = i; j < (size_t)N_NODES * 4; j += stride) { m_n[j] = ninf; sum_n[j] = 0.f; }
  for (size_t j = i; j < (size_t)N_EDGES * 4; j += stride) { m_e[j] = ninf; sum_e[j] = 0.f; }
  for (size_t j = i; j < (size_t)N_NODES * D_OUT; j += stride) x_agg[j] = 0.f;
  for (size_t j = i; j < (size_t)N_EDGES * D_OUT; j += stride) y_agg[j] = 0.f;
}

__device__ __forceinline__ void atomicMaxF(float* addr, float v) {
  if (v >= 0.f) atomicMax((int*)addr, __float_as_int(v));
  else          atomicMin((unsigned int*)addr, __float_as_uint(v));
}
__device__ __forceinline__ float lrelu(float x) { return x >= 0.f ? x : 0.2f * x; }

// pass 1: raw scores per nnz + atomic segment max
__global__ __launch_bounds__(256) void smax_k(const int* __restrict__ node,
                                              const int* __restrict__ edge,
                                              const float* __restrict__ sn,
                                              const float* __restrict__ se,
                                              const float* __restrict__ temp,
                                              float* __restrict__ s_nnz,
                                              float* __restrict__ m_n,
                                              float* __restrict__ m_e) {
  const int i = blockIdx.x * 256 + threadIdx.x;
  if (i >= NNZ) return;
  const int n = node[i], e = edge[i];
  const float invT = 1.0f / temp[0];
  #pragma unroll
  for (int h = 0; h < HEADS; ++h) {
    const float s1 = lrelu(sn[n * 8 + h] + se[e * 8 + h]) * invT;         // e2n: qn + ke
    const float s2 = lrelu(se[e * 8 + 4 + h] + sn[n * 8 + 4 + h]) * invT; // n2e: qe + kn
    s_nnz[(size_t)i * 8 + h] = s1;
    s_nnz[(size_t)i * 8 + 4 + h] = s2;
    atomicMaxF(&m_n[n * 4 + h], s1);
    atomicMaxF(&m_e[e * 4 + h], s2);
  }
}

// pass 2: exp(s - m) in place + atomic segment sum
__global__ __launch_bounds__(256) void ssum_k(const int* __restrict__ node,
                                              const int* __restrict__ edge,
                                              float* __restrict__ s_nnz,
                                              const float* __restrict__ m_n,
                                              const float* __restrict__ m_e,
                                              float* __restrict__ sum_n,
                                              float* __restrict__ sum_e) {
  const int i = blockIdx.x * 256 + threadIdx.x;
  if (i >= NNZ) return;
  const int n = node[i], e = edge[i];
  #pragma unroll
  for (int h = 0; h < HEADS; ++h) {
    const float e1 = __expf(s_nnz[(size_t)i * 8 + h] - m_n[n * 4 + h]);
    const float e2 = __expf(s_nnz[(size_t)i * 8 + 4 + h] - m_e[e * 4 + h]);
    s_nnz[(size_t)i * 8 + h] = e1;
    s_nnz[(size_t)i * 8 + 4 + h] = e2;
    atomicAdd(&sum_n[n * 4 + h], e1);
    atomicAdd(&sum_e[e * 4 + h], e2);
  }
}

// pass 3: per-nnz head-mean attention weights
__global__ __launch_bounds__(256) void sweight_k(const int* __restrict__ node,
                                                 const int* __restrict__ edge,
                                                 const float* __restrict__ s_nnz,
                                                 const float* __restrict__ sum_n,
                                                 const float* __restrict__ sum_e,
                                                 float* __restrict__ w) {
  const int i = blockIdx.x * 256 + threadIdx.x;
  if (i >= NNZ) return;
  const int n = node[i], e = edge[i];
  float w1 = 0.f, w2 = 0.f;
  #pragma unroll
  for (int h = 0; h < HEADS; ++h) {
    w1 += s_nnz[(size_t)i * 8 + h] / sum_n[n * 4 + h];
    w2 += s_nnz[(size_t)i * 8 + 4 + h] / sum_e[e * 4 + h];
  }
  w[(size_t)i * 2 + 0] = 0.25f * w1;
  w[(size_t)i * 2 + 1] = 0.25f * w2;
}

// scatter: one wave per nnz; lane covers 4 columns (float4) in each direction
__global__ __launch_bounds__(256) void scatter_k(const int* __restrict__ node,
                                                 const int* __restrict__ edge,
                                                 const float* __restrict__ w,
                                                 const float* __restrict__ x_trans,
                                                 const float* __restrict__ y_trans,
                                                 float* __restrict__ x_agg,
                                                 float* __restrict__ y_agg) {
  const int wid = (blockIdx.x * 256 + threadIdx.x) >> 5;
  const int lane = threadIdx.x & 31;
  if (wid >= NNZ) return;
  const int n = node[wid], e = edge[wid];
  const float we = w[(size_t)wid * 2 + 0];   // e2n: into x_agg[node] from y_trans[edge]
  const float wn = w[(size_t)wid * 2 + 1];   // n2e: into y_agg[edge] from x_trans[node]
  const float4 yv = ((const float4*)(y_trans + (size_t)e * D_OUT))[lane];
  const float4 xv = ((const float4*)(x_trans + (size_t)n * D_OUT))[lane];
  float* xa = x_agg + (size_t)n * D_OUT + lane * 4;
  float* ya = y_agg + (size_t)e * D_OUT + lane * 4;
  atomicAdd(xa + 0, we * yv.x); atomicAdd(xa + 1, we * yv.y);
  atomicAdd(xa + 2, we * yv.z); atomicAdd(xa + 3, we * yv.w);
  atomicAdd(ya + 0, wn * xv.x); atomicAdd(ya + 1, wn * xv.y);
  atomicAdd(ya + 2, wn * xv.z); atomicAdd(ya + 3, wn * xv.w);
}

// residual + LayerNorm + ReLU; one wave per row (128 cols -> 4 per lane)
__device__ __forceinline__ float waveSum(float v) {
  #pragma unroll
  for (int off = 16; off > 0; off >>= 1) v += __shfl_xor(v, off, 32);
  return v;
}

__global__ __launch_bounds__(256) void ln_relu_k(const float* __restrict__ trans,
                                                 const float* __restrict__ agg,
                                                 const float* __restrict__ g,
                                                 const float* __restrict__ b,
                                                 float* __restrict__ out, int M) {
  const int wid = (blockIdx.x * 256 + threadIdx.x) >> 5;
  const int lane = threadIdx.x & 31;
  if (wid >= M) return;
  const float4 t = ((const float4*)(trans + (size_t)wid * D_OUT))[lane];
  const float4 a = ((const float4*)(agg + (size_t)wid * D_OUT))[lane];
  float h0 = t.x + a.x, h1 = t.y + a.y, h2 = t.z + a.z, h3 = t.w + a.w;
  const float mu = waveSum(h0 + h1 + h2 + h3) * (1.0f / D_OUT);
  h0 -= mu; h1 -= mu; h2 -= mu; h3 -= mu;
  const float var = waveSum(h0 * h0 + h1 * h1 + h2 * h2 + h3 * h3) * (1.0f / D_OUT);
  const float inv = rsqrtf(var + LN_EPS);
  const float4 gv = ((const float4*)g)[lane];
  const float4 bv = ((const float4*)b)[lane];
  float4 o;
  o.x = fmaxf(h0 * inv * gv.x + bv.x, 0.f);
  o.y = fmaxf(h1 * inv * gv.y + bv.y, 0.f);
  o.z = fmaxf(h2 * inv * gv.z + bv.z, 0.f);
  o.w = fmaxf(h3 * inv * gv.w + bv.w, 0.f);
  ((float4*)(out + (size_t)wid * D_OUT))[lane] = o;
}

// ---------------------------------------------------------------------------
extern "C" void kernel_launch(void* const* d_in, const int* in_sizes, int n_in,
                              void* d_out, int out_size, void* d_ws, size_t ws_size,
                              hipStream_t stream) {
  const float* x      = (const float*)d_in[0];
  const float* y      = (const float*)d_in[1];
  const float* Wn     = (const float*)d_in[2];
  const float* bn     = (const float*)d_in[3];
  const float* Wh     = (const float*)d_in[4];
  const float* bh     = (const float*)d_in[5];
  const float* Wnp    = (const float*)d_in[6];
  const float* Whp    = (const float*)d_in[7];
  const float* av_e2n = (const float*)d_in[8];
  const float* av_n2e = (const float*)d_in[9];
  const float* temp   = (const float*)d_in[10];
  const float* ln_n_g = (const float*)d_in[11];
  const float* ln_n_b = (const float*)d_in[12];
  const float* ln_h_g = (const float*)d_in[13];
  const float* ln_h_b = (const float*)d_in[14];
  const int*   hyper  = (const int*)d_in[15];
  const int* node = hyper;
  const int* edge = hyper + NNZ;

  float* ws = (float*)d_ws;
  float* x_trans = ws;                                     // N*128
  float* y_trans = x_trans + (size_t)N_NODES * D_OUT;      // E*128
  float* sn      = y_trans + (size_t)N_EDGES * D_OUT;      // N*8
  float* se      = sn + (size_t)N_NODES * 8;               // E*8
  float* m_n     = se + (size_t)N_EDGES * 8;               // N*4
  float* m_e     = m_n + (size_t)N_NODES * 4;              // E*4
  float* sum_n   = m_e + (size_t)N_EDGES * 4;              // N*4
  float* sum_e   = sum_n + (size_t)N_NODES * 4;            // E*4
  float* x_agg   = sum_e + (size_t)N_EDGES * 4;            // N*128
  float* y_agg   = x_agg + (size_t)N_NODES * D_OUT;        // E*128
  float* s_nnz   = y_agg + (size_t)N_EDGES * D_OUT;        // NNZ*8
  float* wbuf    = s_nnz + (size_t)NNZ * 8;                // NNZ*2
  float* Vn      = wbuf + (size_t)NNZ * 2;                 // 1024
  float* Ve      = Vn + 1024;                              // 1024

  float* out = (float*)d_out;

  // 1-2) feature GEMMs via WMMA
  gemm_wmma_k<<<(N_NODES + 127) / 128, 256, 0, stream>>>(x, Wn, bn, x_trans, N_NODES);
  gemm_wmma_k<<<(N_EDGES + 127) / 128, 256, 0, stream>>>(y, Wh, bh, y_trans, N_EDGES);

  // 3) collapse projection + attention vectors into [128,8] per side
  build_attn_vecs_k<<<8, 256, 0, stream>>>(Wnp, Whp, av_e2n, av_n2e, Vn, Ve);

  // 4-5) per-row q/k scores
  scores_k<<<(N_NODES * 8 + 255) / 256, 256, 0, stream>>>(x_trans, Vn, sn, N_NODES);
  scores_k<<<(N_EDGES * 8 + 255) / 256, 256, 0, stream>>>(y_trans, Ve, se, N_EDGES);

  // 6) init softmax/agg buffers
  init_k<<<2048, 256, 0, stream>>>(m_n, m_e, sum_n, sum_e, x_agg, y_agg);

  // 7-9) segment softmax (max, exp+sum, weights)
  const int nnzBlocks = (NNZ + 255) / 256;
  smax_k<<<nnzBlocks, 256, 0, stream>>>(node, edge, sn, se, temp, s_nnz, m_n, m_e);
  ssum_k<<<nnzBlocks, 256, 0, stream>>>(node, edge, s_nnz, m_n, m_e, sum_n, sum_e);
  sweight_k<<<nnzBlocks, 256, 0, stream>>>(node, edge, s_nnz, sum_n, sum_e, wbuf);

  // 10) weighted scatter-add (wave per nnz)
  scatter_k<<<NNZ / 8, 256, 0, stream>>>(node, edge, wbuf, x_trans, y_trans, x_agg, y_agg);

  // 11-12) residual + LN + ReLU into concatenated output
  ln_relu_k<<<(N_NODES * 32 + 255) / 256, 256, 0, stream>>>(x_trans, x_agg, ln_n_g, ln_n_b,
                                                            out, N_NODES);
  ln_relu_k<<<(N_EDGES * 32 + 255) / 256, 256, 0, stream>>>(y_trans, y_agg, ln_h_g, ln_h_b,
                                                            out + (size_t)N_NODES * D_OUT, N_EDGES);
}